// EquivariantGNN_16664473108518
// MI455X (gfx1250) — compile-verified
//
#include <hip/hip_runtime.h>
#include <cmath>

// ---------------------------------------------------------------------------
// EquivariantGNN forward for MI455X (gfx1250, wave32, WMMA).
// Dense GEMMs use V_WMMA_F32_16X16X4_F32 (fp32 matrix pipe, 16x16 tiles, K=4).
// Edge attention: one wave32 per edge; lane l owns dims [4l,4l+4), head = l>>2.
// ---------------------------------------------------------------------------

typedef float v2f __attribute__((ext_vector_type(2)));
typedef float v8f __attribute__((ext_vector_type(8)));

#define DIM 128   // H*DH = transformer dim
#define NHEAD 8
#define NF 125    // node features from embedding

// ------------------------- helpers -----------------------------------------

__device__ __forceinline__ void atomicMaxFloat(float* addr, float value) {
  // monotonic int-encoding trick; works with -inf (0xFF800000) init.
  if (value >= 0.0f) {
    atomicMax((int*)addr, __float_as_int(value));
  } else {
    atomicMin((unsigned int*)addr, __float_as_uint(value));
  }
}

// ------------------------- init --------------------------------------------

__global__ __launch_bounds__(256) void gnn_init_kernel(
    float* __restrict__ agg, float* __restrict__ mbuf, float* __restrict__ den,
    float* __restrict__ d_out, const float* __restrict__ lin_b, int n) {
  int idx = blockIdx.x * blockDim.x + threadIdx.x;
  int nd = n * DIM;
  int nh = n * NHEAD;
  if (idx < nd) agg[idx] = 0.0f;
  if (idx < nh) {
    mbuf[idx] = __int_as_float(0xFF800000);  // -inf
    den[idx] = 0.0f;
  }
  if (idx == 0) d_out[0] = lin_b[0];
}

// ------------------------- build x = [pos | emb[z]] ------------------------

__global__ __launch_bounds__(256) void gnn_build_x_kernel(
    const float* __restrict__ pos, const float* __restrict__ emb,
    const int* __restrict__ z, float* __restrict__ x, int n) {
  int idx = blockIdx.x * blockDim.x + threadIdx.x;
  if (idx >= n * DIM) return;
  int node = idx >> 7;
  int d = idx & (DIM - 1);
  float val = (d < 3) ? pos[node * 3 + d] : emb[z[node] * NF + (d - 3)];
  x[idx] = val;
}

// ------------------------- WMMA 16x16 tile over K=128 ----------------------
// A fragment (16x4 f32): lanes 0-15 hold K=kk+0,kk+1; lanes 16-31 K=kk+2,kk+3.
// B fragment (4x16 f32): same K split, N = lane&15.
// C/D (16x16 f32): VGPR g <-> row g (lanes 0-15) / row g+8 (lanes 16-31).

__device__ __forceinline__ v8f wmma_tile_128(const float* __restrict__ xs,
                                             const float* __restrict__ W,
                                             int colBase, int lane) {
  const int half = lane >> 4;   // 0 or 1
  const int l15 = lane & 15;
  v8f acc = {};
#pragma unroll
  for (int kk = 0; kk < DIM; kk += 4) {
    v2f a, b;
    const float* ap = xs + l15 * DIM + kk + half * 2;
    a.x = ap[0];
    a.y = ap[1];
    const float* bp = W + (size_t)(kk + half * 2) * DIM + colBase + l15;
    b.x = bp[0];
    b.y = bp[DIM];
    acc = __builtin_amdgcn_wmma_f32_16x16x4_f32(
        /*neg_a=*/false, a, /*neg_b=*/false, b,
        /*c_mod=*/(short)0, acc, /*reuse_a=*/false, /*reuse_b=*/false);
  }
  return acc;
}

// ------------------------- QKV GEMM: q|k|v = x @ {Wq,Wk,Wv} ----------------

__global__ __launch_bounds__(256) void gnn_gemm_qkv_kernel(
    const float* __restrict__ x, const float* __restrict__ Wq,
    const float* __restrict__ Wk, const float* __restrict__ Wv,
    float* __restrict__ qo, float* __restrict__ ko, float* __restrict__ vo,
    int rows) {
  __shared__ float xs[16 * DIM];
  const int tid = threadIdx.x;
  const int rowBase = blockIdx.x * 16;
  for (int i = tid; i < 16 * DIM; i += 256) {
    int r = rowBase + (i >> 7);
    xs[i] = (r < rows) ? x[(size_t)r * DIM + (i & (DIM - 1))] : 0.0f;
  }
  __syncthreads();

  const int wave = tid >> 5;
  const int lane = tid & 31;
  const int colBase = wave * 16;
  const int half = lane >> 4;
  const int l15 = lane & 15;

  const float* Ws[3] = {Wq, Wk, Wv};
  float* Cs[3] = {qo, ko, vo};
#pragma unroll
  for (int m = 0; m < 3; ++m) {
    __builtin_prefetch(Ws[m] + colBase, 0, 0);  // global_prefetch_b8
    v8f acc = wmma_tile_128(xs, Ws[m], colBase, lane);
#pragma unroll
    for (int g = 0; g < 8; ++g) {
      int r = rowBase + g + 8 * half;
      if (r < rows) Cs[m][(size_t)r * DIM + colBase + l15] = acc[g];
    }
  }
}

// ------------------------- out = agg @ Wo + x (residual) -------------------

__global__ __launch_bounds__(256) void gnn_gemm_out_kernel(
    const float* __restrict__ agg, const float* __restrict__ Wo,
    const float* __restrict__ residual, float* __restrict__ out, int rows) {
  __shared__ float xs[16 * DIM];
  const int tid = threadIdx.x;
  const int rowBase = blockIdx.x * 16;
  for (int i = tid; i < 16 * DIM; i += 256) {
    int r = rowBase + (i >> 7);
    xs[i] = (r < rows) ? agg[(size_t)r * DIM + (i & (DIM - 1))] : 0.0f;
  }
  __syncthreads();

  const int wave = tid >> 5;
  const int lane = tid & 31;
  const int colBase = wave * 16;
  const int half = lane >> 4;
  const int l15 = lane & 15;

  v8f acc = wmma_tile_128(xs, Wo, colBase, lane);
#pragma unroll
  for (int g = 0; g < 8; ++g) {
    int r = rowBase + g + 8 * half;
    if (r < rows) {
      size_t o = (size_t)r * DIM + colBase + l15;
      out[o] = acc[g] + residual[o];
    }
  }
}

// ------------------------- edge pass 1: logits + segment max ---------------
// One wave32 per edge. Lane l owns dims [4l, 4l+4); head h = l>>2.

__global__ __launch_bounds__(256) void gnn_edge_logits_kernel(
    const float* __restrict__ q, const float* __restrict__ k,
    const float* __restrict__ edge_attr, const float* __restrict__ We,
    const int* __restrict__ srcA, const int* __restrict__ dstA,
    float* __restrict__ wbuf, float* __restrict__ mbuf, int E) {
  const int e = (blockIdx.x * blockDim.x + threadIdx.x) >> 5;
  const int lane = threadIdx.x & 31;
  if (e >= E) return;
  const int s = srcA[e];
  const int d = dstA[e];
  const int d0 = lane * 4;

  const float4 q4 = *(const float4*)(q + (size_t)d * DIM + d0);
  const float4 k4 = *(const float4*)(k + (size_t)s * DIM + d0);

  const float ea0 = edge_attr[(size_t)e * 4 + 0];
  const float ea1 = edge_attr[(size_t)e * 4 + 1];
  const float ea2 = edge_attr[(size_t)e * 4 + 2];
  const float ea3 = edge_attr[(size_t)e * 4 + 3];
  const float4* We4 = (const float4*)We;  // We[j*128 + d0] -> We4[j*32 + lane]
  const float4 w0 = We4[lane], w1 = We4[32 + lane], w2 = We4[64 + lane],
               w3 = We4[96 + lane];
  float ebx = ea0 * w0.x + ea1 * w1.x + ea2 * w2.x + ea3 * w3.x;
  float eby = ea0 * w0.y + ea1 * w1.y + ea2 * w2.y + ea3 * w3.y;
  float ebz = ea0 * w0.z + ea1 * w1.z + ea2 * w2.z + ea3 * w3.z;
  float ebw = ea0 * w0.w + ea1 * w1.w + ea2 * w2.w + ea3 * w3.w;

  float p = q4.x * (k4.x + ebx) + q4.y * (k4.y + eby) +
            q4.z * (k4.z + ebz) + q4.w * (k4.w + ebw);
  // reduce the 4 lanes of each head
  p += __shfl_xor(p, 1, 32);
  p += __shfl_xor(p, 2, 32);

  if ((lane & 3) == 0) {
    const int h = lane >> 2;
    const float logit = p * 0.25f;  // 1/sqrt(DH=16)
    wbuf[(size_t)e * NHEAD + h] = logit;
    atomicMaxFloat(&mbuf[(size_t)d * NHEAD + h], logit);
  }
}

// ------------------------- edge pass 2: w = exp(l - m), den += w -----------

__global__ __launch_bounds__(256) void gnn_edge_softmax_kernel(
    float* __restrict__ wbuf, const float* __restrict__ mbuf,
    float* __restrict__ den, const int* __restrict__ dstA, int e8) {
  int idx = blockIdx.x * blockDim.x + threadIdx.x;
  if (idx >= e8) return;
  int e = idx >> 3;
  int h = idx & 7;
  int d = dstA[e];
  float w = __expf(wbuf[idx] - mbuf[(size_t)d * NHEAD + h]);
  wbuf[idx] = w;
  atomicAdd(&den[(size_t)d * NHEAD + h], w);
}

// ------------------------- edge pass 3: agg[dst] += alpha * (v[src]+eb) ----

__global__ __launch_bounds__(256) void gnn_edge_agg_kernel(
    const float* __restrict__ v, const float* __restrict__ edge_attr,
    const float* __restrict__ We, const int* __restrict__ srcA,
    const int* __restrict__ dstA, const float* __restrict__ wbuf,
    const float* __restrict__ den, float* __restrict__ agg, int E) {
  const int e = (blockIdx.x * blockDim.x + threadIdx.x) >> 5;
  const int lane = threadIdx.x & 31;
  if (e >= E) return;
  const int s = srcA[e];
  const int d = dstA[e];
  const int h = lane >> 2;
  const float w = wbuf[(size_t)e * NHEAD + h];
  const float dn = den[(size_t)d * NHEAD + h];
  const float alpha = w / (dn + 1e-9f);
  const int d0 = lane * 4;

  const float4 v4 = *(const float4*)(v + (size_t)s * DIM + d0);

  const float ea0 = edge_attr[(size_t)e * 4 + 0];
  const float ea1 = edge_attr[(size_t)e * 4 + 1];
  const float ea2 = edge_attr[(size_t)e * 4 + 2];
  const float ea3 = edge_attr[(size_t)e * 4 + 3];
  const float4* We4 = (const float4*)We;
  const float4 w0 = We4[lane], w1 = We4[32 + lane], w2 = We4[64 + lane],
               w3 = We4[96 + lane];
  float ebx = ea0 * w0.x + ea1 * w1.x + ea2 * w2.x + ea3 * w3.x;
  float eby = ea0 * w0.y + ea1 * w1.y + ea2 * w2.y + ea3 * w3.y;
  float ebz = ea0 * w0.z + ea1 * w1.z + ea2 * w2.z + ea3 * w3.z;
  float ebw = ea0 * w0.w + ea1 * w1.w + ea2 * w2.w + ea3 * w3.w;

  float* ag = agg + (size_t)d * DIM + d0;
  atomicAdd(ag + 0, alpha * (v4.x + ebx));
  atomicAdd(ag + 1, alpha * (v4.y + eby));
  atomicAdd(ag + 2, alpha * (v4.z + ebz));
  atomicAdd(ag + 3, alpha * (v4.w + ebw));
}

// ------------------------- finalize: mean(relu(out) @ lin_w) ---------------

__global__ __launch_bounds__(256) void gnn_finalize_kernel(
    const float* __restrict__ out, const float* __restrict__ lin_w,
    float* __restrict__ d_out, int n) {
  const int node = blockIdx.x * blockDim.x + threadIdx.x;
  float y = 0.0f;
  if (node < n) {
    const float4* o = (const float4*)(out + (size_t)node * DIM);
    const float4* lw = (const float4*)lin_w;
#pragma unroll
    for (int i = 0; i < DIM / 4; ++i) {
      float4 a = o[i];
      float4 b = lw[i];
      y += fmaxf(a.x, 0.0f) * b.x + fmaxf(a.y, 0.0f) * b.y +
           fmaxf(a.z, 0.0f) * b.z + fmaxf(a.w, 0.0f) * b.w;
    }
  }
  __shared__ float red[256];
  red[threadIdx.x] = y;
  __syncthreads();
  for (int s = 128; s > 0; s >>= 1) {
    if (threadIdx.x < s) red[threadIdx.x] += red[threadIdx.x + s];
    __syncthreads();
  }
  if (threadIdx.x == 0) atomicAdd(d_out, red[0] / (float)n);
}

// ------------------------- launcher ----------------------------------------

extern "C" void kernel_launch(void* const* d_in, const int* in_sizes, int n_in,
                              void* d_out, int out_size, void* d_ws,
                              size_t ws_size, hipStream_t stream) {
  const float* pos       = (const float*)d_in[0];
  const float* edge_attr = (const float*)d_in[1];
  const float* emb       = (const float*)d_in[2];
  const float* Wq        = (const float*)d_in[3];
  const float* Wk        = (const float*)d_in[4];
  const float* Wv        = (const float*)d_in[5];
  const float* We        = (const float*)d_in[6];
  const float* Wo        = (const float*)d_in[7];
  const float* lin_w     = (const float*)d_in[8];
  const float* lin_b     = (const float*)d_in[9];
  const int*   z         = (const int*)d_in[10];
  const int*   edge_index= (const int*)d_in[11];

  const int n = in_sizes[0] / 3;   // nodes
  const int e = in_sizes[1] / 4;   // edges
  const int* srcA = edge_index;
  const int* dstA = edge_index + e;

  float* ws = (float*)d_ws;
  const size_t nd = (size_t)n * DIM;
  float* x    = ws;
  float* q    = ws + nd;
  float* k    = ws + 2 * nd;
  float* v    = ws + 3 * nd;
  float* agg  = ws + 4 * nd;
  float* wbuf = ws + 5 * nd;
  float* mbuf = wbuf + (size_t)e * NHEAD;
  float* den  = mbuf + (size_t)n * NHEAD;
  float* out  = q;  // q is dead after the edge passes; reuse for the output
  float* dout = (float*)d_out;

  const dim3 blk(256);
  const int nbND = (int)((nd + 255) / 256);
  const int rowBlocks = (n + 15) / 16;
  const int edgeBlocks = (e + 7) / 8;  // 8 waves (edges) per 256-thread block
  const int e8 = e * NHEAD;

  gnn_init_kernel<<<nbND, blk, 0, stream>>>(agg, mbuf, den, dout, lin_b, n);
  gnn_build_x_kernel<<<nbND, blk, 0, stream>>>(pos, emb, z, x, n);
  gnn_gemm_qkv_kernel<<<rowBlocks, blk, 0, stream>>>(x, Wq, Wk, Wv, q, k, v, n);
  gnn_edge_logits_kernel<<<edgeBlocks, blk, 0, stream>>>(q, k, edge_attr, We,
                                                         srcA, dstA, wbuf,
                                                         mbuf, e);
  gnn_edge_softmax_kernel<<<(e8 + 255) / 256, blk, 0, stream>>>(wbuf, mbuf,
                                                                den, dstA, e8);
  gnn_edge_agg_kernel<<<edgeBlocks, blk, 0, stream>>>(v, edge_attr, We, srcA,
                                                      dstA, wbuf, den, agg, e);
  gnn_gemm_out_kernel<<<rowBlocks, blk, 0, stream>>>(agg, Wo, x, out, n);
  gnn_finalize_kernel<<<(n + 255) / 256, blk, 0, stream>>>(out, lin_w, dout, n);
}